// BasePropagationGraphPositionalEncoding_69449621176714
// MI455X (gfx1250) — compile-verified
//
#include <hip/hip_runtime.h>

// CDNA5 / gfx1250: wave32, WMMA (not MFMA).
typedef __attribute__((ext_vector_type(2))) float v2f;
typedef __attribute__((ext_vector_type(4))) float v4f;
typedef __attribute__((ext_vector_type(8))) float v8f;

#define EPS_   0.01f
#define B_     8
#define K_     9
#define N_     1024
#define D_     64
#define TM_    16           // output rows per workgroup tile
#define MC_    64           // m-chunk streamed per iteration
#define WROW_  (MC_ + 4)    // padded LDS stride (68): keeps 16B alignment, spreads banks
#define XROW_  (D_ + 4)     // padded LDS stride (68)
#define NTHREADS_ 128       // 4 x wave32

// Workgroup barrier that publishes LDS only: waits DScnt==0 (LDS stores visible)
// but deliberately leaves LOADcnt outstanding so software-pipelined global
// loads keep flying across the barrier. (__syncthreads' fence drains loadcnt
// to 0, which would serialize the pipeline -- seen in round-2 asm.)
__device__ __forceinline__ void wg_barrier_lds()
{
    asm volatile(
        "s_wait_dscnt 0x0\n\t"
        "s_barrier_signal -1\n\t"
        "s_barrier_wait -1"
        ::: "memory");
}

// emb[b,n,d] = (1/S) * sum_m [ sum_k (1-eps)^k * gr[b,k,n,m] ] * x[b,m,d]
__global__ __launch_bounds__(NTHREADS_) void
gpe_wmma_f32_kernel(const float* __restrict__ x,
                    const float* __restrict__ gr,
                    float* __restrict__ out)
{
    __shared__ float w_lds[TM_ * WROW_];   // 16 x 68 f32 = 4352 B  (weighted kernel tile)
    __shared__ float x_lds[MC_ * XROW_];   // 64 x 68 f32 = 17408 B (x chunk)

    const int wg   = blockIdx.x;               // 0 .. B_*(N_/TM_)-1
    const int b    = wg / (N_ / TM_);
    const int n0   = (wg % (N_ / TM_)) * TM_;

    const int tid  = threadIdx.x;              // 0..127
    const int wave = tid >> 5;                 // 0..3 (wave32)
    const int lane = tid & 31;
    const int half = lane >> 4;                // 0: lanes 0-15, 1: lanes 16-31
    const int l16  = lane & 15;

    // Normalizer: S = sum_{k=0}^{K-1} (1-eps)^k  (constant-folds at -O3)
    float S = 0.f, c = 1.f;
#pragma unroll
    for (int k = 0; k < K_; ++k) { S += c; c *= (1.f - EPS_); }
    const float invS = 1.f / S;

    const float* grb = gr + (size_t)b * K_ * N_ * N_;   // [K, N, N]
    const float* xb  = x  + (size_t)b * N_ * D_;        // [N, D]

    // Streaming-phase roles: row r (16 rows), 8 threads per row, 8 f32 each.
    const int r  = tid >> 3;            // 0..15
    const int c8 = (tid & 7) * 8;       // 0,8,...,56

    // x-staging roles: 2 threads per m-row, 32 consecutive f32 each.
    const int xr = tid >> 1;            // 0..63
    const int xc = (tid & 1) * 32;      // 0 or 32

    const float* gr_base = grb + (size_t)(n0 + r) * N_ + c8;   // + m0 + k*N*N
    const float* x_base  = xb + (size_t)xr * D_ + xc;          // + m0*D

    // Each wave accumulates one 16-wide d-slice of the 16x64 output tile.
    const int d0 = wave * 16;
    v8f acc = {0.f, 0.f, 0.f, 0.f, 0.f, 0.f, 0.f, 0.f};

    // ---- Software pipeline: raw chunk lives in registers one iteration ahead ----
    v4f g0v[K_], g1v[K_];   // gr chunk: 9 k-planes x 8 f32 per thread
    v4f xv[8];              // x chunk: 32 f32 per thread

    {   // preload chunk m0 = 0 (coalesced b128 streams)
        const float* p = gr_base;
#pragma unroll
        for (int k = 0; k < K_; ++k) {
            g0v[k] = *(const v4f*)(p);
            g1v[k] = *(const v4f*)(p + 4);
            p += (size_t)N_ * N_;
        }
#pragma unroll
        for (int j = 0; j < 8; ++j) xv[j] = *(const v4f*)(x_base + j * 4);
    }

    for (int m0 = 0; m0 < N_; m0 += MC_) {
        // ---- Phase 1: K-weighted reduction of the in-register chunk ----
        float wacc[8];
#pragma unroll
        for (int j = 0; j < 8; ++j) wacc[j] = 0.f;
        float ck = 1.0f;
#pragma unroll
        for (int k = 0; k < K_; ++k) {
            wacc[0] = fmaf(ck, g0v[k].x, wacc[0]);
            wacc[1] = fmaf(ck, g0v[k].y, wacc[1]);
            wacc[2] = fmaf(ck, g0v[k].z, wacc[2]);
            wacc[3] = fmaf(ck, g0v[k].w, wacc[3]);
            wacc[4] = fmaf(ck, g1v[k].x, wacc[4]);
            wacc[5] = fmaf(ck, g1v[k].y, wacc[5]);
            wacc[6] = fmaf(ck, g1v[k].z, wacc[6]);
            wacc[7] = fmaf(ck, g1v[k].w, wacc[7]);
            ck *= (1.f - EPS_);
        }
        // 16B-aligned LDS staging -> ds_store_b128
        {
            float* pw = &w_lds[r * WROW_ + c8];
            v4f wv0 = {wacc[0], wacc[1], wacc[2], wacc[3]};
            v4f wv1 = {wacc[4], wacc[5], wacc[6], wacc[7]};
            *(v4f*)(pw)     = wv0;
            *(v4f*)(pw + 4) = wv1;
#pragma unroll
            for (int j = 0; j < 8; ++j)
                *(v4f*)(&x_lds[xr * XROW_ + xc + j * 4]) = xv[j];
        }

        // ---- Issue next chunk's global loads NOW: with the LDS-only barrier
        //      below, their latency overlaps the barrier + 16-WMMA phase.
        if (m0 + MC_ < N_) {
            const float* p = gr_base + (m0 + MC_);
#pragma unroll
            for (int k = 0; k < K_; ++k) {
                g0v[k] = *(const v4f*)(p);
                g1v[k] = *(const v4f*)(p + 4);
                p += (size_t)N_ * N_;
            }
            const float* px = x_base + (size_t)(m0 + MC_) * D_;
#pragma unroll
            for (int j = 0; j < 8; ++j) xv[j] = *(const v4f*)(px + j * 4);
        }
        // ---- L2 prefetch two chunks ahead (global_prefetch_b8): extra HBM
        //      latency slack beyond the one-chunk register pipeline.
        if (m0 + 2 * MC_ < N_) {
            const float* pp = gr_base + (m0 + 2 * MC_);
#pragma unroll
            for (int k = 0; k < K_; ++k) {
                __builtin_prefetch(pp, 0, 0);
                pp += (size_t)N_ * N_;
            }
            __builtin_prefetch(x_base + (size_t)(m0 + 2 * MC_) * D_, 0, 0);
        }

        wg_barrier_lds();   // publish LDS tiles; global loads stay in flight

        // ---- Phase 2: 16 WMMA steps of K=4 over this chunk ----
        // A 16x4 f32 layout: lanes 0-15 hold (row=l16, K=0,1); lanes 16-31 (row=l16, K=2,3)
        // B 4x16  f32 layout: lanes 0-15 hold (K=0,1, col=l16); lanes 16-31 (K=2,3, col=l16)
#pragma unroll
        for (int mm = 0; mm < MC_ / 4; ++mm) {
            const float* pa = &w_lds[l16 * WROW_ + mm * 4 + half * 2]; // 8B aligned -> ds_load_b64
            v2f av = *(const v2f*)pa;

            const float* pb = &x_lds[(mm * 4 + half * 2) * XROW_ + d0 + l16];
            v2f bv; bv.x = pb[0]; bv.y = pb[XROW_];

            // D = A*B + C   (f32 16x16x4)
            acc = __builtin_amdgcn_wmma_f32_16x16x4_f32(
                /*neg_a=*/false, av, /*neg_b=*/false, bv,
                /*c_mod=*/(short)0, acc, /*reuse_a=*/false, /*reuse_b=*/false);
        }

        wg_barrier_lds();   // protect LDS tiles before next iteration's stores
    }

    // ---- Epilogue: C/D layout -> out[b, n0+row, d0+col], scaled by 1/S ----
#pragma unroll
    for (int j = 0; j < 8; ++j) {
        const int row = n0 + j + half * 8;
        out[((size_t)b * N_ + row) * D_ + d0 + l16] = acc[j] * invS;
    }
}

extern "C" void kernel_launch(void* const* d_in, const int* in_sizes, int n_in,
                              void* d_out, int out_size, void* d_ws, size_t ws_size,
                              hipStream_t stream)
{
    const float* x  = (const float*)d_in[0];   // [B, N, D]
    const float* gr = (const float*)d_in[1];   // [B, K, N, N]
    float* out = (float*)d_out;                // [B, N, D]

    dim3 grid(B_ * (N_ / TM_));                // 512 workgroups
    dim3 block(NTHREADS_);                     // 4 x wave32
    gpe_wmma_f32_kernel<<<grid, block, 0, stream>>>(x, gr, out);
}